// TokenKVBuilder_13812614824506
// MI455X (gfx1250) — compile-verified
//
#include <hip/hip_runtime.h>
#include <math.h>

// ---------------------------------------------------------------------------
// TokenKVBuilder for MI455X (gfx1250): pure bandwidth op (~1.07 GB traffic,
// ~50us roofline at 23.3 TB/s). No matmul -> no WMMA. Strategy:
//   * 2D grid (S, B): no integer-division SALU blob for block decomposition
//   * b128 vector loads of gathered table rows (block-uniform token -> s_load)
//   * RoPE angles on the fly (one v_exp_f32 + v_sin/v_cos per pair-of-pairs)
//   * non-temporal b128 stores for all outputs (keep L2 for table-row reuse)
//   * v-path copied via gfx1250 async LDS DMA:
//       global_load_async_to_lds_b128 -> s_wait_asynccnt 0
//       -> global_store_async_from_lds_b128  (ASYNCcnt-tracked)
//     issued before the k RoPE compute so the compute hides its latency.
// ---------------------------------------------------------------------------

typedef __attribute__((ext_vector_type(4))) float v4f;

#define D_HEAD 128
#define HQ     32
#define HKV    8
// -ln(10000)/64  (inv_freq(p) = exp(p * this))
#define NEG_LN_BASE_OVER_HALF_D (-0.14391156831212783f)
// 10000^(-1/64) == exp(NEG_LN_BASE_OVER_HALF_D): inv_freq(p+1) = inv_freq(p)*this
#define INV_FREQ_STEP            (0.8659643233600653f)

__device__ __forceinline__ v4f rope4(v4f x, float pos, int d) {
    // d is the in-head offset of x.x (multiple of 4): pairs (d,d+1),(d+2,d+3)
    float p0 = (float)(d >> 1);
    float f0 = __expf(p0 * NEG_LN_BASE_OVER_HALF_D);
    float f1 = f0 * INV_FREQ_STEP;          // saves a second v_exp_f32
    float s0, c0, s1, c1;
    __sincosf(pos * f0, &s0, &c0);
    __sincosf(pos * f1, &s1, &c1);
    v4f r;
    r.x = x.x * c0 - x.y * s0;
    r.y = x.y * c0 + x.x * s0;
    r.z = x.z * c1 - x.w * s1;
    r.w = x.w * c1 + x.z * s1;
    return r;
}

// Grid (S, B): one block per (b, s) token. 256 threads, one float4 per thread
// for each of k (RoPE'd, through VGPRs) and v (copy, through async LDS DMA).
__global__ __launch_bounds__(256) void kv_gather_rope_kernel(
    const int*   __restrict__ ctx,  // (B, S)
    const float* __restrict__ kt,   // (V, HKV*D)
    const float* __restrict__ vt,   // (V, HKV*D)
    float*       __restrict__ ok,   // (B, HKV, S, D)
    float*       __restrict__ ov,   // (B, HKV, S, D)
    int S)
{
    __shared__ float vbuf[HKV * D_HEAD];  // 4 KB staging for the v row

    const int s   = blockIdx.x;
    const int b   = blockIdx.y;
    const int tok = ctx[(size_t)b * S + s];   // block-uniform -> scalar load

    const int tid = threadIdx.x;          // 0..255
    const int f   = tid << 2;             // flat offset in 1024-float row
    const int h   = f >> 7;               // head index
    const int d   = f & (D_HEAD - 1);     // in-head offset (multiple of 4)

    const size_t row   = (size_t)tok * (HKV * D_HEAD);
    const size_t obase = (((size_t)b * HKV + h) * (size_t)S + s) * D_HEAD + d;

    // ---- v path: kick off async DMA global -> LDS (16B per lane) ----------
    const float* vsrc = vt + row + f;
    unsigned lds_off  = (unsigned)(unsigned long long)(&vbuf[f]);
    asm volatile("global_load_async_to_lds_b128 %0, %1, off"
                 :: "v"(lds_off), "v"(vsrc) : "memory");

    // ---- k path: load, RoPE, non-temporal store (hides the async latency) -
    v4f kv = *reinterpret_cast<const v4f*>(kt + row + f);
    v4f ko = rope4(kv, (float)s, d);
    __builtin_nontemporal_store(ko, reinterpret_cast<v4f*>(ok + obase));

    // ---- v path: wait for LDS data, then async DMA LDS -> global ----------
    asm volatile("s_wait_asynccnt 0" ::: "memory");
    float* vdst = ov + obase;
    asm volatile("global_store_async_from_lds_b128 %0, %1, off"
                 :: "v"(vdst), "v"(lds_off) : "memory");
    // s_endpgm performs an implicit wait-idle, draining the async store.
}

// q: B*HQ*D floats total. Grid (4, B): each block covers 1024 floats of the
// gathered 4096-float row, one float4 per thread, RoPE at position S.
__global__ __launch_bounds__(256) void q_gather_rope_kernel(
    const int*   __restrict__ nxt,  // (B,)
    const float* __restrict__ qt,   // (V, HQ*D)
    float*       __restrict__ oq,   // (B, HQ, 1, D) == flat (B, HQ*D)
    int S)
{
    const int chunk = blockIdx.x;       // 0..3  (HQ*D / 1024 == 4)
    const int b     = blockIdx.y;
    const int tok   = nxt[b];

    const int tid = threadIdx.x;
    const int f   = (chunk << 10) + (tid << 2);   // offset in 4096-float row
    const int d   = f & (D_HEAD - 1);

    v4f q = *reinterpret_cast<const v4f*>(qt + (size_t)tok * (HQ * D_HEAD) + f);
    v4f r = rope4(q, (float)S, d);   // q position is S
    __builtin_nontemporal_store(
        r, reinterpret_cast<v4f*>(oq + (size_t)b * (HQ * D_HEAD) + f));
}

extern "C" void kernel_launch(void* const* d_in, const int* in_sizes, int n_in,
                              void* d_out, int out_size, void* d_ws, size_t ws_size,
                              hipStream_t stream) {
    const int*   ctx = (const int*)  d_in[0];   // context_tokens (B,S)
    const int*   nxt = (const int*)  d_in[1];   // next_tokens (B,)
    const float* qt  = (const float*)d_in[2];   // q_table (V, HQ*D)
    const float* kt  = (const float*)d_in[3];   // k_table (V, HKV*D)
    const float* vt  = (const float*)d_in[4];   // v_table (V, HKV*D)
    float* out = (float*)d_out;

    const int B = in_sizes[1];
    const int S = in_sizes[0] / B;

    // d_out = [ q (B*HQ*D) | k (B*HKV*S*D) | v (B*HKV*S*D) ]
    float* oq = out;
    float* ok = out + (size_t)B * HQ * D_HEAD;
    float* ov = ok + (size_t)B * HKV * (size_t)S * D_HEAD;

    kv_gather_rope_kernel<<<dim3(S, B), 256, 0, stream>>>(ctx, kt, vt, ok, ov, S);
    q_gather_rope_kernel<<<dim3(HQ * D_HEAD / 1024, B), 256, 0, stream>>>(nxt, qt, oq, S);
}